// Gate_12103217840540
// MI455X (gfx1250) — compile-verified
//
#include <hip/hip_runtime.h>

// MoE router gate for MI455X (gfx1250, wave32).
//   logits = x[T,2048] @ W[64,2048]^T ; top-6 indices + pre-softmax logits.
// Softmax is monotonic -> skipped (same top-k indices/order as reference).
// GEMM via V_WMMA_F32_16X16X4_F32 (fp32 keeps parity with fp32 reference).
// Staging via GLOBAL_LOAD_ASYNC_TO_LDS_B128 + double-buffered LDS so the
// tensor-core stream overlaps the HBM stream (kernel is HBM-bound: 256 MB of
// x at 23.3 TB/s vs 8.6 GFLOP).

typedef __attribute__((ext_vector_type(2))) float v2f;
typedef __attribute__((ext_vector_type(8))) float v8f;
typedef __attribute__((ext_vector_type(4))) int   v4i;

namespace {
constexpr int kD      = 2048;  // hidden dim
constexpr int kE      = 64;    // experts
constexpr int kKC     = 64;    // K-chunk staged in LDS
constexpr int kMRows  = 32;    // rows per block
constexpr int kStride = 68;    // LDS row stride (floats): (4r+k)%64 bank map ->
                               // conflict-free b64 fragment reads, 16B aligned
constexpr int kAct    = 6;     // top-k
constexpr float kRouteScale = 1.0f;
}

#if __has_builtin(__builtin_amdgcn_global_load_async_to_lds_b128)
#define HAVE_ASYNC_LDS 1
typedef __attribute__((address_space(1))) v4i as1_v4i;
typedef __attribute__((address_space(3))) v4i as3_v4i;

__device__ __forceinline__ void async_copy16(const float* gsrc, float* lds) {
  // 16 bytes per lane, global -> LDS, no VGPR round trip (ASYNCcnt tracked).
  __builtin_amdgcn_global_load_async_to_lds_b128(
      (as1_v4i*)(void*)gsrc, (as3_v4i*)(void*)lds, /*offset=*/0, /*cpol=*/0);
}
__device__ __forceinline__ void wait_async0() {
#if __has_builtin(__builtin_amdgcn_s_wait_asynccnt)
  __builtin_amdgcn_s_wait_asynccnt(0);
#else
  asm volatile("s_wait_asynccnt 0x0" ::: "memory");
#endif
}
#else
#define HAVE_ASYNC_LDS 0
#endif

// 16 WMMAs over one staged K-chunk for this wave's 16x16 tile.
__device__ __forceinline__ void wmma_chunk(const float* __restrict__ xrow,
                                           const float* __restrict__ wrow,
                                           v8f& c) {
#pragma unroll
  for (int kk = 0; kk < kKC; kk += 4) {
    v2f a = *(const v2f*)(xrow + kk);
    v2f b = *(const v2f*)(wrow + kk);
    c = __builtin_amdgcn_wmma_f32_16x16x4_f32(
        /*neg_a=*/false, a, /*neg_b=*/false, b,
        /*c_mod=*/(short)0, c, /*reuse_a=*/false, /*reuse_b=*/false);
  }
}

__global__ __launch_bounds__(256) void gate_wmma_kernel(
    const float* __restrict__ x, const float* __restrict__ W,
    int* __restrict__ idx_out, float* __restrict__ w_out) {
#if HAVE_ASYNC_LDS
  __shared__ float xs[2][kMRows * kStride];  // ping-pong x tiles
  __shared__ float ws[2][kE * kStride];      // ping-pong W tiles
#else
  __shared__ float xs1[kMRows * kStride];
  __shared__ float ws1[kE * kStride];
#endif
  __shared__ float lgs[kMRows * kE];  // logits [32][64]

  const int tid   = threadIdx.x;
  const int lane  = tid & 31;
  const int wave  = tid >> 5;
  const int half  = lane >> 4;  // 0: lanes 0-15, 1: lanes 16-31
  const int r     = lane & 15;
  const int mTile = wave >> 2;  // rows 16*mTile..
  const int nTile = wave & 3;   // experts 16*nTile..
  const long rowBase = (long)blockIdx.x * kMRows;

  // Per-thread staging assignment (float4 units): coalesced 256B per 16 lanes.
  const int xrow0 = tid >> 4, xc4 = tid & 15;  // + j*16 rows (j=0..1)
  v8f c = {};

#if HAVE_ASYNC_LDS
  // ---- async double-buffered pipeline -------------------------------------
  {
#pragma unroll
    for (int j = 0; j < 2; ++j)
      async_copy16(x + (rowBase + xrow0 + j * 16) * kD + xc4 * 4,
                   &xs[0][(xrow0 + j * 16) * kStride + xc4 * 4]);
#pragma unroll
    for (int j = 0; j < 4; ++j)
      async_copy16(W + (long)(xrow0 + j * 16) * kD + xc4 * 4,
                   &ws[0][(xrow0 + j * 16) * kStride + xc4 * 4]);
  }
  int cur = 0;
  for (int k0 = 0; k0 < kD; k0 += kKC) {
    wait_async0();     // my copies into buf[cur] landed
    __syncthreads();   // everyone's copies landed / everyone done reading nxt
    if (k0 + kKC < kD) {
      const int nxt = cur ^ 1;
#pragma unroll
      for (int j = 0; j < 2; ++j)
        async_copy16(x + (rowBase + xrow0 + j * 16) * kD + (k0 + kKC) + xc4 * 4,
                     &xs[nxt][(xrow0 + j * 16) * kStride + xc4 * 4]);
#pragma unroll
      for (int j = 0; j < 4; ++j)
        async_copy16(W + (long)(xrow0 + j * 16) * kD + (k0 + kKC) + xc4 * 4,
                     &ws[nxt][(xrow0 + j * 16) * kStride + xc4 * 4]);
    }
    wmma_chunk(&xs[cur][(mTile * 16 + r) * kStride + 2 * half],
               &ws[cur][(nTile * 16 + r) * kStride + 2 * half], c);
    cur ^= 1;
  }
#else
  // ---- fallback: batched register staging, single buffer ------------------
  for (int k0 = 0; k0 < kD; k0 += kKC) {
    __syncthreads();
    float4 vx[2], vw[4];
#pragma unroll
    for (int j = 0; j < 2; ++j)
      vx[j] = *(const float4*)(x + (rowBase + xrow0 + j * 16) * kD + k0 + xc4 * 4);
#pragma unroll
    for (int j = 0; j < 4; ++j)
      vw[j] = *(const float4*)(W + (long)(xrow0 + j * 16) * kD + k0 + xc4 * 4);
#pragma unroll
    for (int j = 0; j < 2; ++j)
      *(float4*)(&xs1[(xrow0 + j * 16) * kStride + xc4 * 4]) = vx[j];
#pragma unroll
    for (int j = 0; j < 4; ++j)
      *(float4*)(&ws1[(xrow0 + j * 16) * kStride + xc4 * 4]) = vw[j];
    __syncthreads();
    wmma_chunk(&xs1[(mTile * 16 + r) * kStride + 2 * half],
               &ws1[(nTile * 16 + r) * kStride + 2 * half], c);
  }
#endif

  // Scatter C tile to LDS logits: vgpr i -> M = i + 8*half, N = r.
#pragma unroll
  for (int i = 0; i < 8; ++i) {
    lgs[(mTile * 16 + i + 8 * half) * kE + nTile * 16 + r] = c[i];
  }
  __syncthreads();

  // Top-6 per row (each wave handles 4 rows). 2 experts per lane, then a
  // 5-step butterfly argmax; tie -> lower index (matches jax.lax.top_k).
  for (int q = 0; q < 4; ++q) {
    const int row = wave * 4 + q;
    float v0 = lgs[row * kE + lane];
    float v1 = lgs[row * kE + lane + 32];
    int i0 = lane, i1 = lane + 32;
    const long t = rowBase + row;
#pragma unroll
    for (int j = 0; j < kAct; ++j) {
      float bv;
      int bi;
      if (v0 >= v1) { bv = v0; bi = i0; } else { bv = v1; bi = i1; }
#pragma unroll
      for (int off = 16; off >= 1; off >>= 1) {
        const float ov = __shfl_xor(bv, off, 32);
        const int oi = __shfl_xor(bi, off, 32);
        if (ov > bv || (ov == bv && oi < bi)) { bv = ov; bi = oi; }
      }
      if (lane == 0) {
        idx_out[t * kAct + j] = bi;              // int32 indices
        w_out[t * kAct + j] = bv * kRouteScale;  // original logits
      }
      if (bi == i0) v0 = -INFINITY;
      if (bi == i1) v1 = -INFINITY;
    }
  }
}

extern "C" void kernel_launch(void* const* d_in, const int* in_sizes, int n_in,
                              void* d_out, int out_size, void* d_ws,
                              size_t ws_size, hipStream_t stream) {
  const float* x = (const float*)d_in[0];
  const float* W = (const float*)d_in[1];
  const int T = in_sizes[0] / kD;  // 32768

  // Tuple output, flat in return order: [T*6] int32 indices, [T*6] f32 weights.
  int* idx_out = (int*)d_out;
  float* w_out = (float*)d_out + (size_t)T * kAct;

  dim3 grid(T / kMRows), block(256);
  hipLaunchKernelGGL(gate_wmma_kernel, grid, block, 0, stream, x, W, idx_out,
                     w_out);
}